// HBVModel_60679297958216
// MI455X (gfx1250) — compile-verified
//
#include <hip/hip_runtime.h>
#include <math.h>

#define WMAX 1024                 // max supported t_max+1 (actual here: 25)
#define WEXT_CENTER WMAX          // wext[WEXT_CENTER + k] == w[k], zero-padded around
#define WEXT_SIZE   (2 * WMAX + 64)

typedef __attribute__((ext_vector_type(2))) float v2f;
typedef __attribute__((ext_vector_type(8))) float v8f;

// ---------------------------------------------------------------------------
// Kernel 1: normalized gamma unit-hydrograph weights, written into a
// zero-padded table wext so the conv kernel needs no range checks:
//   wext[WEXT_CENTER + k] = w[k] for k in [0, t_max], 0 in a generous margin.
// ---------------------------------------------------------------------------
__global__ void hbv_weights_kernel(const float* __restrict__ ga,
                                   const float* __restrict__ gt,
                                   float* __restrict__ wext)
{
    if (threadIdx.x != 0 || blockIdx.x != 0) return;
    float a   = ga[0];
    float tau = gt[0];
    int t_max = (int)(3.0f * a * tau);
    if (t_max > WMAX - 1) t_max = WMAX - 1;
    int Lw = t_max + 1;

    // zero margin covering every index the conv can touch: k in [-16, PAD+48)
    int lo = WEXT_CENTER - 64;
    int hi = WEXT_CENTER + t_max + 64;
    if (hi > WEXT_SIZE) hi = WEXT_SIZE;
    for (int i = lo; i < hi; ++i) wext[i] = 0.0f;

    float denom = expf(lgammaf(a)) * powf(tau, a);
    float s = 0.0f;
    for (int i = 0; i < Lw; ++i) {
        float ts = (float)(i + 1);
        float wi = powf(ts, a - 1.0f) * expf(-ts / tau) / denom;
        wext[WEXT_CENTER + i] = wi;
        s += wi;
    }
    float inv = 1.0f / s;
    for (int i = 0; i < Lw; ++i) wext[WEXT_CENTER + i] *= inv;
}

// ---------------------------------------------------------------------------
// Kernel 2: sequential HBV recurrence, one thread per basin.
// Writes Q time-major [T, B] so conv loads/stores coalesce.
// ---------------------------------------------------------------------------
__global__ void hbv_scan_kernel(const float* __restrict__ P,
                                const int*   __restrict__ lengths,
                                const float* ETP, const float* iSM,
                                const float* iSUZ, const float* iSLZ,
                                const float* FC, const float* Lparm,
                                const float* LP, const float* beta,
                                const float* K0, const float* K1,
                                const float* Kd, const float* K2,
                                float* __restrict__ Qws, int Bn, int Tn)
{
    int b = blockIdx.x * blockDim.x + threadIdx.x;
    if (b >= Bn) return;

    const float etp0 = ETP[0];
    const float fc = FC[0], l = Lparm[0], lp = LP[0], bexp = beta[0];
    const float k0 = K0[0], k1 = K1[0], kd = Kd[0], k2 = K2[0];
    const float inv_fclp = 1.0f / (fc * lp);
    const float inv_fc   = 1.0f / fc;

    float SM  = iSM[0];
    float SUZ = iSUZ[0];
    float SLZ = iSLZ[0];

    int len = lengths[b];
    if (len > Tn) len = Tn;
    const float* __restrict__ Prow = P + (size_t)b * (size_t)Tn;

    int t = 0;
    for (; t < len; ++t) {
        if ((t & 63) == 0) __builtin_prefetch(Prow + t + 512, 0, 1);
        float Pt = Prow[t];
        float ET = etp0 * fminf(SM * inv_fclp, 1.0f);
        float R  = __powf(SM * inv_fc, bexp);            // SM >= 0 always
        SM = fmaxf(SM + (Pt - ET - R), 0.0f);
        float Qd   = kd * SUZ;
        float SUZn = fmaxf(SUZ + (R - k0 * fmaxf(SUZ - l, 0.0f) - k1 * SUZ - Qd), 0.0f);
        float SLZn = fmaxf(SLZ + (Qd - k2 * SLZ), 0.0f);
        SUZ = SUZn;
        SLZ = SLZn;
        float Q = k0 * fmaxf(SUZ - l, 0.0f) + k1 * SUZ + k2 * SLZ;
        Qws[(size_t)t * (size_t)Bn + b] = Q;
    }
    for (; t < Tn; ++t) Qws[(size_t)t * (size_t)Bn + b] = 0.0f;
}

// ---------------------------------------------------------------------------
// WMMA helpers. A layout (ISA 7.12.2, 32-bit 16x4): lane<16 holds K=p0,p0+1
// in v0/v1 with p0=4c; lane>=16 holds K=p0+2,p0+3. B (4x16): VGPR r holds
// rows K=r (lanes 0-15) and K=2+r (lanes 16-31), col = lane&15.
// ---------------------------------------------------------------------------
__device__ __forceinline__ v8f wmma_f32(v2f a, v2f b, v8f acc)
{
    return __builtin_amdgcn_wmma_f32_16x16x4_f32(
        false, a, false, b, (short)0, acc, false, false);
}

// Fast interior path: no bounds checks, fully unrolled -> CHUNKS static WMMAs.
// Qb = Qws + (t0-PAD)*Bn + b0 + m (this lane's column), not offset by hi (added via p0 inside).
// wc = wext + WEXT_CENTER + PAD + m  (band: W[p][m] = wc[-p], zero-padded).
template <int CHUNKS>
__device__ __forceinline__ v8f conv_tile_fast(const float* __restrict__ Qb,
                                              const float* __restrict__ wc,
                                              int Bn, int hi)
{
    v8f acc = {};
    #pragma unroll
    for (int c = 0; c < CHUNKS; ++c) {
        int p0 = (c << 2) + (hi << 1);
        v2f a = { Qb[(size_t)p0 * Bn], Qb[(size_t)(p0 + 1) * Bn] };
        v2f b = { wc[-p0], wc[-(p0 + 1)] };
        acc = wmma_f32(a, b, acc);
    }
    return acc;
}

// ---------------------------------------------------------------------------
// Kernel 3: causal FIR conv as f32 WMMA band matmul + epilogue.
// Grid: x = time-tile waves (8 waves/block), y = batch tile (no int division).
// ---------------------------------------------------------------------------
__global__ void hbv_conv_wmma_kernel(const float* __restrict__ Qws,
                                     const float* __restrict__ wext,
                                     const float* __restrict__ ga,
                                     const float* __restrict__ gt,
                                     const float* __restrict__ Area,
                                     const float* __restrict__ baseflow,
                                     const int*   __restrict__ lengths,
                                     float* __restrict__ out,
                                     int Bn, int Tn, int tilesT)
{
    int lane = threadIdx.x & 31;
    int tt   = blockIdx.x * (blockDim.x >> 5) + (threadIdx.x >> 5);
    if (tt >= tilesT) return;               // wave-uniform exit, EXEC stays full
    int t0 = tt << 4;
    int b0 = blockIdx.y << 4;

    int t_max = (int)(3.0f * ga[0] * gt[0]);
    if (t_max > WMAX - 1) t_max = WMAX - 1;
    int PAD    = (t_max + 3) & ~3;          // window left-pad, multiple of 4
    int chunks = (PAD + 16) >> 2;

    int m  = lane & 15;
    int hi = lane >> 4;
    const float* wc = wext + WEXT_CENTER + PAD + m;

    v8f acc;
    if (chunks == 10 && t0 >= PAD && (t0 + 16) <= Tn && (b0 + 16) <= Bn) {
        // interior tile, t_max in [21,24] (the actual case): 10 unrolled WMMAs
        const float* Qb = Qws + (size_t)(t0 - PAD) * (size_t)Bn + b0 + m;
        acc = conv_tile_fast<10>(Qb, wc, Bn, hi);
    } else {
        bool rowOK = (b0 + m) < Bn;
        v8f a2 = {};
        for (int c = 0; c < chunks; ++c) {
            int p0 = (c << 2) + (hi << 1);
            int tA = t0 - PAD + p0;
            float ax = (rowOK && tA >= 0 && tA < Tn)
                         ? Qws[(size_t)tA * Bn + b0 + m] : 0.0f;
            float ay = (rowOK && (tA + 1) >= 0 && (tA + 1) < Tn)
                         ? Qws[(size_t)(tA + 1) * Bn + b0 + m] : 0.0f;
            v2f a = {ax, ay};
            v2f b = {wc[-p0], wc[-(p0 + 1)]};   // zero-padded, no clamp needed
            a2 = wmma_f32(a, b, a2);
        }
        acc = a2;
    }

    // Epilogue: C/D layout -> lane, vgpr r: row M = r + 8*hi, col N = m
    int tcol = t0 + m;
    if (tcol < Tn) {
        #pragma unroll
        for (int r = 0; r < 8; ++r) {
            int brow = b0 + r + (hi << 3);
            if (brow < Bn) {
                float val = acc[r] * (Area[brow] * (1.0f / 3.6f)) + baseflow[brow];
                val = (tcol < lengths[brow]) ? val : 0.0f;
                out[(size_t)brow * (size_t)Tn + tcol] = val;
            }
        }
    }
}

// ---------------------------------------------------------------------------
extern "C" void kernel_launch(void* const* d_in, const int* in_sizes, int n_in,
                              void* d_out, int out_size, void* d_ws, size_t ws_size,
                              hipStream_t stream)
{
    const float* ETP      = (const float*)d_in[0];
    const float* P        = (const float*)d_in[1];
    const int*   lengths  = (const int*)  d_in[2];
    const float* iSM      = (const float*)d_in[3];
    const float* iSUZ     = (const float*)d_in[4];
    const float* iSLZ     = (const float*)d_in[5];
    const float* Area     = (const float*)d_in[6];
    const float* baseflow = (const float*)d_in[7];
    const float* FC       = (const float*)d_in[8];
    const float* Lparm    = (const float*)d_in[9];
    const float* LP       = (const float*)d_in[10];
    const float* beta     = (const float*)d_in[11];
    const float* K0       = (const float*)d_in[12];
    const float* K1       = (const float*)d_in[13];
    const float* Kd       = (const float*)d_in[14];
    const float* K2       = (const float*)d_in[15];
    const float* ga       = (const float*)d_in[16];
    const float* gt       = (const float*)d_in[17];
    float* out = (float*)d_out;

    int Bn = in_sizes[6];                  // Area has B elements
    int Tn = in_sizes[1] / Bn;             // P is [B,T,1]

    float* ws   = (float*)d_ws;
    float* wext = ws;                      // WEXT_SIZE floats
    float* Qws  = ws + WEXT_SIZE + 64;     // Q time-major [T, B]

    // 1) gamma weights (device-side: t_max depends on device values)
    hbv_weights_kernel<<<dim3(1), dim3(1), 0, stream>>>(ga, gt, wext);

    // 2) sequential scan: one wave32 per block to spread waves across WGPs
    hbv_scan_kernel<<<dim3((Bn + 31) / 32), dim3(32), 0, stream>>>(
        P, lengths, ETP, iSM, iSUZ, iSLZ, FC, Lparm, LP, beta,
        K0, K1, Kd, K2, Qws, Bn, Tn);

    // 3) WMMA band-matmul convolution: one wave per 16x16 tile,
    //    grid.x = time tiles / 8 waves per block, grid.y = batch tiles
    int tilesB = (Bn + 15) / 16;
    int tilesT = (Tn + 15) / 16;
    int wavesPerBlock = 8;                 // 256 threads
    int blocksX = (tilesT + wavesPerBlock - 1) / wavesPerBlock;
    hbv_conv_wmma_kernel<<<dim3(blocksX, tilesB), dim3(wavesPerBlock * 32), 0, stream>>>(
        Qws, wext, ga, gt, Area, baseflow, lengths, out, Bn, Tn, tilesT);
}